// PlackettLuceLoss_61031485276231
// MI455X (gfx1250) — compile-verified
//
#include <hip/hip_runtime.h>

// Plackett-Luce loss for MI455X (gfx1250).
// Bandwidth-bound: 256 MB read / launch -> ~11 us at 23.3 TB/s HBM.
// Strategy: stage each 8KB score row into LDS with CDNA5 async global->LDS
// loads (ASYNCcnt) while the permutation indices stream into VGPRs (LOADcnt),
// gather from LDS, and compute the reverse cumulative logsumexp as
//   K*M + log(reverse-cumsum(exp(s - M)))
// with a wave32 shuffle scan + cross-wave LDS combine.

#define K_DIM 2048
#define TPB   256
#define EPT   (K_DIM / TPB)   // 8 elements per thread
#define NWAVES (TPB / 32)     // 8 waves per block (wave32)

#if defined(__gfx1250__) && __has_builtin(__builtin_amdgcn_global_load_async_to_lds_b128)
#define HAVE_ASYNC_LDS 1
#else
#define HAVE_ASYNC_LDS 0
#endif

#if HAVE_ASYNC_LDS
typedef __attribute__((ext_vector_type(4))) int v4i_t;
typedef __attribute__((address_space(1))) v4i_t* g_v4i_p;   // global (__device__)
typedef __attribute__((address_space(3))) v4i_t* l_v4i_p;   // LDS (__shared__)
#endif

__global__ __launch_bounds__(TPB) void pl_row_kernel(
    const float* __restrict__ scores,
    const int*   __restrict__ order,
    float*       __restrict__ row_logprob)
{
    __shared__ float s_lds[K_DIM];     // staged score row (8 KB)
    __shared__ float redA[NWAVES];
    __shared__ float redB[NWAVES];
    __shared__ float wtot[NWAVES];

    const int b    = blockIdx.x;
    const int t    = threadIdx.x;
    const int lane = t & 31;
    const int wave = t >> 5;

    const float* srow = scores + (size_t)b * K_DIM;
    const int*   irow = order  + (size_t)b * K_DIM;

    int idx[EPT];

#if HAVE_ASYNC_LDS
    // ---- Issue async stage of the score row into LDS (ASYNCcnt) ----
    // 2048 floats = 512 x b128; 2 async b128 loads per thread.
    __builtin_amdgcn_global_load_async_to_lds_b128(
        (g_v4i_p)(srow + t * 4),
        (l_v4i_p)&s_lds[t * 4],
        0, 0);
    __builtin_amdgcn_global_load_async_to_lds_b128(
        (g_v4i_p)(srow + (t + TPB) * 4),
        (l_v4i_p)&s_lds[(t + TPB) * 4],
        0, 0);

    // ---- Overlap: stream permutation indices into VGPRs (LOADcnt) ----
    {
        const int4* p = reinterpret_cast<const int4*>(irow + t * EPT);
        int4 a = p[0], c = p[1];
        idx[0] = a.x; idx[1] = a.y; idx[2] = a.z; idx[3] = a.w;
        idx[4] = c.x; idx[5] = c.y; idx[6] = c.z; idx[7] = c.w;
    }

    // Wait only for this wave's async LDS writes, then barrier so every
    // wave sees the full row in LDS. Index loads stay in flight past this.
    asm volatile("s_wait_asynccnt 0" ::: "memory");
    __syncthreads();   // B1
#else
    {
        #pragma unroll
        for (int i = 0; i < 2; ++i) {
            int i4 = t + i * TPB;
            float4 v = reinterpret_cast<const float4*>(srow)[i4];
            reinterpret_cast<float4*>(s_lds)[i4] = v;
        }
    }
    {
        const int4* p = reinterpret_cast<const int4*>(irow + t * EPT);
        int4 a = p[0], c = p[1];
        idx[0] = a.x; idx[1] = a.y; idx[2] = a.z; idx[3] = a.w;
        idx[4] = c.x; idx[5] = c.y; idx[6] = c.z; idx[7] = c.w;
    }
    __syncthreads();   // B1
#endif

    // ---- LDS gather + local max / sum ----
    float sv[EPT];
    float smax = -3.402823466e38f;
    float ssum = 0.0f;
    #pragma unroll
    for (int k = 0; k < EPT; ++k) {
        float v = s_lds[idx[k]];
        sv[k] = v;
        smax = fmaxf(smax, v);
        ssum += v;
    }

    // ---- Block max M ----
    #pragma unroll
    for (int d = 16; d >= 1; d >>= 1)
        smax = fmaxf(smax, __shfl_xor(smax, d, 32));
    if (lane == 0) redA[wave] = smax;
    __syncthreads();   // B2
    float M = redA[0];
    #pragma unroll
    for (int w = 1; w < NWAVES; ++w) M = fmaxf(M, redA[w]);

    // ---- exp(s - M) and in-thread reverse suffix sums ----
    float suf[EPT];
    float run = 0.0f;
    #pragma unroll
    for (int k = EPT - 1; k >= 0; --k) {
        run += __expf(sv[k] - M);
        suf[k] = run;
    }
    const float T = run;   // this thread's chunk total

    // ---- Wave32 reverse inclusive scan of chunk totals ----
    float incl = T;
    #pragma unroll
    for (int d = 1; d < 32; d <<= 1) {
        float v = __shfl_down(incl, d, 32);
        if (lane + d < 32) incl += v;
    }
    if (lane == 0) wtot[wave] = incl;   // lane 0 holds the full wave sum
    __syncthreads();   // B3

    float woff = 0.0f;
    #pragma unroll
    for (int w = NWAVES - 1; w >= 0; --w)
        if (w > wave) woff += wtot[w];

    // Exclusive reverse offset for this thread's chunk:
    const float base = woff + (incl - T);

    // ---- sum of log(reverse cumsum) over this thread's elements ----
    float slog = 0.0f;
    #pragma unroll
    for (int k = 0; k < EPT; ++k)
        slog += __logf(base + suf[k]);

    // ---- Block reduce slog and ssum ----
    #pragma unroll
    for (int d = 16; d >= 1; d >>= 1) {
        slog += __shfl_xor(slog, d, 32);
        ssum += __shfl_xor(ssum, d, 32);
    }
    if (lane == 0) { redA[wave] = slog; redB[wave] = ssum; }
    __syncthreads();   // B4

    if (t == 0) {
        float tl = 0.0f, ts = 0.0f;
        #pragma unroll
        for (int w = 0; w < NWAVES; ++w) { tl += redA[w]; ts += redB[w]; }
        // log_prob[b] = sum(s) - K*M - sum_i log(c_i)
        row_logprob[b] = ts - (float)K_DIM * M - tl;
    }
}

__global__ __launch_bounds__(256) void pl_finalize_kernel(
    const float* __restrict__ row_logprob, float* __restrict__ out, int nrows)
{
    __shared__ double red[256];
    double s = 0.0;
    for (int i = threadIdx.x; i < nrows; i += 256)
        s += (double)row_logprob[i];
    red[threadIdx.x] = s;
    __syncthreads();
    for (int stride = 128; stride > 0; stride >>= 1) {
        if (threadIdx.x < stride) red[threadIdx.x] += red[threadIdx.x + stride];
        __syncthreads();
    }
    if (threadIdx.x == 0)
        out[0] = (float)(-red[0] / (double)nrows);
}

extern "C" void kernel_launch(void* const* d_in, const int* in_sizes, int n_in,
                              void* d_out, int out_size, void* d_ws, size_t ws_size,
                              hipStream_t stream) {
    const float* scores = (const float*)d_in[0];
    const int*   order  = (const int*)d_in[1];   // JAX x64-off: int64 -> int32
    float* out    = (float*)d_out;
    float* rowbuf = (float*)d_ws;                // B floats of scratch

    const int nrows = in_sizes[0] / K_DIM;       // 16384

    pl_row_kernel<<<nrows, TPB, 0, stream>>>(scores, order, rowbuf);
    pl_finalize_kernel<<<1, 256, 0, stream>>>(rowbuf, out, nrows);
}